// Graph_Predictor_29540785062522
// MI455X (gfx1250) — compile-verified
//
#include <hip/hip_runtime.h>

typedef __attribute__((ext_vector_type(2))) float v2f;
typedef __attribute__((ext_vector_type(8))) float v8f;

#define NNODES 50000
#define NEDGES 600000
#define DHID   128
#define DOUT   64
#define NGRAPH 64

// ---------------- utility ----------------
__global__ void zero_f32(float* __restrict__ p, int n) {
  int i = blockIdx.x * blockDim.x + threadIdx.x;
  if (i < n) p[i] = 0.0f;
}

// ---------------- edge aggregation ----------------
__global__ void edge_deg_kernel(const int* __restrict__ dst, float* __restrict__ deg, int E) {
  int e = blockIdx.x * blockDim.x + threadIdx.x;
  if (e < E) atomicAdd(&deg[dst[e]], 1.0f);
}

// one float4 chunk per thread: idx spans E*32 (128 feats / 4)
__global__ void edge_scatter_kernel(const float* __restrict__ x,
                                    const int* __restrict__ src,
                                    const int* __restrict__ dst,
                                    float* __restrict__ agg, int E) {
  int idx = blockIdx.x * blockDim.x + threadIdx.x;
  int e = idx >> 5;
  if (e >= E) return;
  int c = (idx & 31) * 4;
  const float4 v = *(const float4*)(x + (size_t)src[e] * DHID + c);
  float* a = agg + (size_t)dst[e] * DHID + c;
  atomicAdd(a + 0, v.x);
  atomicAdd(a + 1, v.y);
  atomicAdd(a + 2, v.z);
  atomicAdd(a + 3, v.w);
}

__global__ void normalize_agg_kernel(float* __restrict__ agg,
                                     const float* __restrict__ deg, int N) {
  int idx = blockIdx.x * blockDim.x + threadIdx.x;
  int n = idx >> 5;
  if (n >= N) return;
  int c = (idx & 31) * 4;
  float d = deg[n];
  d = (d < 1.0f) ? 1.0f : d;
  float4* p = (float4*)(agg + (size_t)n * DHID + c);
  float4 v = *p;
  v.x /= d; v.y /= d; v.z /= d; v.w /= d;
  *p = v;
}

// ---------------- fused WMMA GEMM ----------------
// Software-pipelined K-loop for one operand pair:
// four 16x16 accumulators (row tiles) share each B fragment; next k-step's
// A/B fragments are loaded BEFORE the current WMMAs so the s_wait for them
// hides behind 4 matrix-pipe ops. Fully unrolled (K=128 -> 32 steps) so the
// prefetch guard folds at compile time.
template <int DOUTC>
__device__ __forceinline__ void gemm_phase(const float* __restrict__ X,
                                           const float* __restrict__ sW,
                                           int arow0, int arow1, int arow2, int arow3,
                                           int kA, int colIdx,
                                           v8f& c0, v8f& c1, v8f& c2, v8f& c3) {
  const float* a0 = X + (size_t)arow0 * DHID + kA;
  const float* a1 = X + (size_t)arow1 * DHID + kA;
  const float* a2 = X + (size_t)arow2 * DHID + kA;
  const float* a3 = X + (size_t)arow3 * DHID + kA;
  const float* bb = sW + colIdx * 4 + kA;   // 8B aligned (kA even)

  // prime the pipeline
  v2f b   = *(const v2f*)(bb);
  v2f av0 = *(const v2f*)(a0);
  v2f av1 = *(const v2f*)(a1);
  v2f av2 = *(const v2f*)(a2);
  v2f av3 = *(const v2f*)(a3);

#pragma unroll
  for (int k = 0; k < DHID; k += 4) {
    v2f bn = b, an0 = av0, an1 = av1, an2 = av2, an3 = av3;
    if (k + 4 < DHID) {                    // folds away after full unroll
      bn  = *(const v2f*)(bb + ((k + 4) >> 2) * (DOUTC * 4));
      an0 = *(const v2f*)(a0 + k + 4);
      an1 = *(const v2f*)(a1 + k + 4);
      an2 = *(const v2f*)(a2 + k + 4);
      an3 = *(const v2f*)(a3 + k + 4);
    }
    c0 = __builtin_amdgcn_wmma_f32_16x16x4_f32(false, av0, false, b, (short)0, c0, false, false);
    c1 = __builtin_amdgcn_wmma_f32_16x16x4_f32(false, av1, false, b, (short)0, c1, false, false);
    c2 = __builtin_amdgcn_wmma_f32_16x16x4_f32(false, av2, false, b, (short)0, c2, false, false);
    c3 = __builtin_amdgcn_wmma_f32_16x16x4_f32(false, av3, false, b, (short)0, c3, false, false);
    b = bn; av0 = an0; av1 = an1; av2 = an2; av3 = an3;
  }
}

// Y[N x DOUTC] = act( X1[N x 128] @ W1[128 x DOUTC] (+ X2 @ W2) + bias )
// Block: (DOUTC/16) waves; block covers 64 rows (4 row tiles per wave).
// W staged in LDS in k-interleaved layout [K/4][DOUTC][4] -> the K-pair a lane
// needs is contiguous (ds_load_b64 / ds_load_2addr_stride64_b64).
//
// V_WMMA_F32_16X16X4_F32 operand layout (ISA 7.12.2):
//   A (16x4):  lane l holds row M=l&15; lanes<16 -> K={0,1}, lanes>=16 -> K={2,3}
//   B (4x16):  lane l holds col N=l&15; lanes<16 -> K={0,1}, lanes>=16 -> K={2,3}
//   C/D:       lane l holds col N=l&15, rows M=(l>>4)*8 + vgpr(0..7)
template <int DOUTC, bool HASX2, bool RELU>
__global__ void sage_wmma_gemm(const float* __restrict__ X1, const float* __restrict__ W1,
                               const float* __restrict__ X2, const float* __restrict__ W2,
                               const float* __restrict__ bias, float* __restrict__ Y,
                               int N) {
  extern __shared__ float smem[];
  float* sW1 = smem;
  float* sW2 = smem + DHID * DOUTC;

  const int tid = threadIdx.x;
  for (int i = tid; i < DHID * DOUTC; i += blockDim.x) {
    int k = i / DOUTC, n = i % DOUTC;               // DOUTC power of 2 -> shifts
    sW1[(k >> 2) * (DOUTC * 4) + n * 4 + (k & 3)] = W1[i];
  }
  if (HASX2) {
    for (int i = tid; i < DHID * DOUTC; i += blockDim.x) {
      int k = i / DOUTC, n = i % DOUTC;
      sW2[(k >> 2) * (DOUTC * 4) + n * 4 + (k & 3)] = W2[i];
    }
  }
  __syncthreads();

  const int wave = tid >> 5;
  const int lane = tid & 31;
  const int col0 = wave * 16;
  const int n16  = lane & 15;
  const int hi   = lane >> 4;   // lane-half selects K sub-pair
  const int kA   = hi * 2;      // K offset for this lane-half: 0 or 2
  const int colIdx = col0 + n16;

  const int rowBase = blockIdx.x * 64;
  // Clamp tail rows (EXEC must stay all-1s for WMMA); stores predicated below.
  int arow0 = rowBase + 0  + n16; if (arow0 >= N) arow0 = N - 1;
  int arow1 = rowBase + 16 + n16; if (arow1 >= N) arow1 = N - 1;
  int arow2 = rowBase + 32 + n16; if (arow2 >= N) arow2 = N - 1;
  int arow3 = rowBase + 48 + n16; if (arow3 >= N) arow3 = N - 1;

  v8f c0 = {0.f,0.f,0.f,0.f,0.f,0.f,0.f,0.f};
  v8f c1 = c0, c2 = c0, c3 = c0;

  gemm_phase<DOUTC>(X1, sW1, arow0, arow1, arow2, arow3, kA, colIdx, c0, c1, c2, c3);
  if (HASX2) {
    gemm_phase<DOUTC>(X2, sW2, arow0, arow1, arow2, arow3, kA, colIdx, c0, c1, c2, c3);
  }

  // ---- bias + activation + predicated store ----
  const int mbase = hi * 8;
  const float bv = bias[colIdx];
  v8f acc[4] = {c0, c1, c2, c3};
#pragma unroll
  for (int rt = 0; rt < 4; ++rt) {
    const int row0 = rowBase + rt * 16;
#pragma unroll
    for (int r = 0; r < 8; ++r) {
      int row = row0 + mbase + r;
      if (row < N) {
        float v = acc[rt][r] + bv;
        if (RELU) v = (v > 0.0f) ? v : 0.0f;
        Y[(size_t)row * DOUTC + colIdx] = v;
      }
    }
  }
}

// ---------------- graph pooling ----------------
__global__ void pool_scatter_kernel(const float* __restrict__ x64,
                                    const int* __restrict__ gid,
                                    float* __restrict__ out,
                                    float* __restrict__ cnt, int N) {
  int idx = blockIdx.x * blockDim.x + threadIdx.x;   // N * 64
  if (idx >= N * DOUT) return;
  int node = idx >> 6;
  int f = idx & 63;
  int g = gid[node];
  atomicAdd(&out[g * DOUT + f], x64[idx]);
  if (f == 0) atomicAdd(&cnt[g], 1.0f);
}

__global__ void pool_div_kernel(float* __restrict__ out, const float* __restrict__ cnt) {
  int idx = blockIdx.x * blockDim.x + threadIdx.x;   // 64*64
  if (idx >= NGRAPH * DOUT) return;
  float d = cnt[idx >> 6];
  d = (d < 1.0f) ? 1.0f : d;
  out[idx] /= d;
}

// ---------------- driver ----------------
static inline int cdiv(int a, int b) { return (a + b - 1) / b; }

extern "C" void kernel_launch(void* const* d_in, const int* in_sizes, int n_in,
                              void* d_out, int out_size, void* d_ws, size_t ws_size,
                              hipStream_t stream) {
  const float* h        = (const float*)d_in[0];
  const int*   edge_src = (const int*)d_in[1];
  const int*   edge_dst = (const int*)d_in[2];
  const int*   graph_id = (const int*)d_in[3];
  const float* ws0 = (const float*)d_in[4];
  const float* wn0 = (const float*)d_in[5];
  const float* bs0 = (const float*)d_in[6];
  const float* ws1 = (const float*)d_in[7];
  const float* wn1 = (const float*)d_in[8];
  const float* bs1 = (const float*)d_in[9];
  const float* ws2 = (const float*)d_in[10];
  const float* wn2 = (const float*)d_in[11];
  const float* bs2 = (const float*)d_in[12];
  const float* lw0 = (const float*)d_in[13];
  const float* lb0 = (const float*)d_in[14];
  const float* lw1 = (const float*)d_in[15];
  const float* lb1 = (const float*)d_in[16];
  const float* lw2 = (const float*)d_in[17];
  const float* lb2 = (const float*)d_in[18];
  float* out = (float*)d_out;

  const int ND = NNODES * DHID;            // 6.4M floats
  float* agg  = (float*)d_ws;              // also reused as final [N x 64]
  float* bufA = agg  + ND;
  float* bufB = bufA + ND;
  float* deg  = bufB + ND;
  float* cnt  = deg + NNODES;

  const int Z256  = 256;
  const int gZero = cdiv(ND, Z256);
  const int gDeg  = cdiv(NEDGES, Z256);
  const int gScat = cdiv(NEDGES * 32, Z256);
  const int gNorm = cdiv(NNODES * 32, Z256);
  const int gGemm = cdiv(NNODES, 64);
  const size_t smem2 = (size_t)2 * DHID * DHID * sizeof(float);   // 128 KB (two W)
  const size_t smem1 = (size_t)DHID * DHID * sizeof(float);       // 64 KB
  const size_t smemP = (size_t)DHID * DOUT * sizeof(float);       // 32 KB

  // degree depends only on edge structure: compute once
  zero_f32<<<cdiv(NNODES, Z256), Z256, 0, stream>>>(deg, NNODES);
  edge_deg_kernel<<<gDeg, Z256, 0, stream>>>(edge_dst, deg, NEDGES);

  // ---- SAGE layer 0: bufA = relu(h@ws0 + bs0 + mean@wn0) ----
  zero_f32<<<gZero, Z256, 0, stream>>>(agg, ND);
  edge_scatter_kernel<<<gScat, Z256, 0, stream>>>(h, edge_src, edge_dst, agg, NEDGES);
  normalize_agg_kernel<<<gNorm, Z256, 0, stream>>>(agg, deg, NNODES);
  sage_wmma_gemm<DHID, true, true><<<gGemm, 256, smem2, stream>>>(
      h, ws0, agg, wn0, bs0, bufA, NNODES);

  // ---- SAGE layer 1: bufB = relu(bufA@ws1 + bs1 + mean@wn1) ----
  zero_f32<<<gZero, Z256, 0, stream>>>(agg, ND);
  edge_scatter_kernel<<<gScat, Z256, 0, stream>>>(bufA, edge_src, edge_dst, agg, NEDGES);
  normalize_agg_kernel<<<gNorm, Z256, 0, stream>>>(agg, deg, NNODES);
  sage_wmma_gemm<DHID, true, true><<<gGemm, 256, smem2, stream>>>(
      bufA, ws1, agg, wn1, bs1, bufB, NNODES);

  // ---- SAGE layer 2: bufA = relu(bufB@ws2 + bs2 + mean@wn2) ----
  zero_f32<<<gZero, Z256, 0, stream>>>(agg, ND);
  edge_scatter_kernel<<<gScat, Z256, 0, stream>>>(bufB, edge_src, edge_dst, agg, NEDGES);
  normalize_agg_kernel<<<gNorm, Z256, 0, stream>>>(agg, deg, NNODES);
  sage_wmma_gemm<DHID, true, true><<<gGemm, 256, smem2, stream>>>(
      bufB, ws2, agg, wn2, bs2, bufA, NNODES);

  // ---- MLP ----
  sage_wmma_gemm<DHID, false, true><<<gGemm, 256, smem1, stream>>>(
      bufA, lw0, nullptr, nullptr, lb0, bufB, NNODES);
  sage_wmma_gemm<DHID, false, true><<<gGemm, 256, smem1, stream>>>(
      bufB, lw1, nullptr, nullptr, lb1, bufA, NNODES);
  // projection to 64 (agg reused as [N x 64] output), 4 waves per block
  sage_wmma_gemm<DOUT, false, false><<<gGemm, 128, smemP, stream>>>(
      bufA, lw2, nullptr, nullptr, lb2, agg, NNODES);

  // ---- per-graph mean pool ----
  zero_f32<<<cdiv(NGRAPH * DOUT, Z256), Z256, 0, stream>>>(out, NGRAPH * DOUT);
  zero_f32<<<1, NGRAPH, 0, stream>>>(cnt, NGRAPH);
  pool_scatter_kernel<<<cdiv(NNODES * DOUT, Z256), Z256, 0, stream>>>(agg, graph_id,
                                                                      out, cnt, NNODES);
  pool_div_kernel<<<cdiv(NGRAPH * DOUT, Z256), Z256, 0, stream>>>(out, cnt);
}